// FilterLayer_79980880986848
// MI455X (gfx1250) — compile-verified
//
#include <hip/hip_runtime.h>
#include <math.h>

// ---------------------------------------------------------------------------
// FilterLayer: out[b,c,n,:] = (M_n + I) @ x[b,c,n,:]  (seq len 12, 207 nodes)
//   M_n derived from complex_weight via rfft/irfft (ortho) algebra.
// Memory-bound: 651 MB traffic -> ~28us floor @ 23.3 TB/s.
// Matrix math done with V_WMMA_F32_16X16X4_F32 (3 chunks of K=4).
// ---------------------------------------------------------------------------

typedef float v2f __attribute__((ext_vector_type(2)));
typedef float v8f __attribute__((ext_vector_type(8)));

#define NUM_NODES 207
#define SEQ 12
#define CHANNELS 32

// Precompute G[n][s][t] = M_n[t][s] + (s==t), stored t-fastest (12x12 per node).
__global__ void build_G_kernel(const float* __restrict__ cw, float* __restrict__ G) {
    const int n   = blockIdx.x;        // 0..206
    const int tid = threadIdx.x;       // 0..143
    const int t = tid / SEQ;
    const int s = tid % SEQ;
    const float* wn = cw + n * 14;     // 7 freqs x (re, im)

    const int d = t - s;
    float acc = wn[0];                                        // k = 0 (DC, real part)
    acc += (((t + s) & 1) ? -wn[12] : wn[12]);                // k = 6 (Nyquist, real part)
#pragma unroll
    for (int k = 1; k <= 5; ++k) {
        const float ang = (float)M_PI * (float)(k * d) * (1.0f / 6.0f);
        acc += 2.0f * (wn[2 * k] * cosf(ang) - wn[2 * k + 1] * sinf(ang));
    }
    const float g = acc * (1.0f / 12.0f) + ((t == s) ? 1.0f : 0.0f);
    G[n * 144 + s * SEQ + t] = g;
}

// One wave per 16-row tile: rows = 16 consecutive channels of one (batch, node).
// D(16x16) = X(16x12) * G_n(12x16-padded), via 3x V_WMMA_F32_16X16X4_F32.
__global__ __launch_bounds__(256) void filter_wmma_kernel(
        const float* __restrict__ in, const float* __restrict__ G,
        float* __restrict__ out, int ntiles) {
    const int wavesPerBlock = blockDim.x >> 5;
    const int wave = blockIdx.x * wavesPerBlock + (threadIdx.x >> 5);
    if (wave >= ntiles) return;                 // wave-uniform; EXEC stays all-1s
    const int lane = threadIdx.x & 31;

    // tile -> (b, n, half)
    const int b    = wave / (NUM_NODES * 2);
    const int rem  = wave - b * (NUM_NODES * 2);
    const int n    = rem >> 1;
    const int c0   = (rem & 1) << 4;            // 0 or 16

    const int hi   = lane >> 4;                 // 0: lanes 0-15, 1: lanes 16-31
    const int Ma   = lane & 15;                 // A-matrix row (channel within tile)
    const int koff = hi << 1;                   // A/B K sub-offset {0,2}
    const int t    = lane & 15;                 // B/D column (output time index)
    const int tc   = (t < SEQ) ? t : (SEQ - 1); // clamped column for safe loads
    const float bmask = (t < SEQ) ? 1.0f : 0.0f;

    const size_t rowStride = (size_t)NUM_NODES * SEQ;                  // 2484 floats
    const size_t aBase = ((size_t)(b * CHANNELS + c0 + Ma) * NUM_NODES + n) * SEQ;
    const float* gn = G + n * 144;

    v8f acc = {};
#pragma unroll
    for (int j = 0; j < 3; ++j) {
        const int s0 = 4 * j + koff;            // even -> 8-byte aligned
        const v2f a = *(const v2f*)(in + aBase + s0);
        v2f bf;
        bf.x = gn[s0 * SEQ + tc] * bmask;       // G[s0][t]
        bf.y = gn[(s0 + 1) * SEQ + tc] * bmask; // G[s0+1][t]
        // (neg_a, A, neg_b, B, c_mod, C, reuse_a, reuse_b)
        acc = __builtin_amdgcn_wmma_f32_16x16x4_f32(
            false, a, false, bf, (short)0, acc, false, false);
    }

    // D layout: lane L, vgpr v -> row M = 8*(L>>4)+v, col N = L&15
    if (t < SEQ) {
        const size_t oBase = ((size_t)(b * CHANNELS + c0) * NUM_NODES + n) * SEQ + t;
        const int m0 = hi << 3;
#pragma unroll
        for (int v = 0; v < 8; ++v) {
            out[oBase + (size_t)(m0 + v) * rowStride] = acc[v];
        }
    }
}

extern "C" void kernel_launch(void* const* d_in, const int* in_sizes, int n_in,
                              void* d_out, int out_size, void* d_ws, size_t ws_size,
                              hipStream_t stream) {
    const float* in = (const float*)d_in[0];   // [B,32,207,12] f32
    const float* cw = (const float*)d_in[1];   // [1,207,7,2]   f32
    float* out = (float*)d_out;
    float* G   = (float*)d_ws;                 // 207*144 floats = 119 KB

    const int total = in_sizes[0];
    const int B = total / (CHANNELS * NUM_NODES * SEQ);

    build_G_kernel<<<NUM_NODES, 144, 0, stream>>>(cw, G);

    const int ntiles = B * NUM_NODES * (CHANNELS / 16);   // 16 rows per wave
    const int wavesPerBlock = 8;                           // 256 threads
    const int blocks = (ntiles + wavesPerBlock - 1) / wavesPerBlock;
    filter_wmma_kernel<<<blocks, 256, 0, stream>>>(in, G, out, ntiles);
}